// DVLFN_53575422051006
// MI455X (gfx1250) — compile-verified
//
#include <hip/hip_runtime.h>
#include <math.h>

#define BATCH 128
#define LTOK  256
#define RREG  36
#define RPAD  48
#define DTXT  768
#define DIMG  2048
#define DSOC  10
#define FEAT  50
#define FPAD  64
#define EPSV     0.0025f     /* blur^2 = 0.05^2 */
#define INV_EPS  400.0f
#define NITER    20
#define GAMMA_   0.01f
#define NEG_INF  (-__builtin_inff())

typedef __attribute__((ext_vector_type(2))) float v2f;
typedef __attribute__((ext_vector_type(8))) float v8f;

// fp32 WMMA: D(16x16) = A(16x4) * B(4x16) + C ; exact fp32 like the reference.
__device__ __forceinline__ v8f wmma4(v2f a, v2f b, v8f c) {
  return __builtin_amdgcn_wmma_f32_16x16x4_f32(
      /*neg_a=*/false, a, /*neg_b=*/false, b,
      /*c_mod=*/(short)0, c, /*reuse_a=*/false, /*reuse_b=*/false);
}

#define V8F_ZERO {0.f, 0.f, 0.f, 0.f, 0.f, 0.f, 0.f, 0.f}

__device__ __forceinline__ float block_reduce_sum_256(float v, float* red) {
  int tid = threadIdx.x;
  red[tid] = v;
  __syncthreads();
  for (int s = 128; s > 0; s >>= 1) {
    if (tid < s) red[tid] += red[tid + s];
    __syncthreads();
  }
  float r = red[0];
  __syncthreads();
  return r;
}

// ---------------------------------------------------------------------------
// Pre-swizzle W [K,50] into WMMA B-fragment order, zero-padded to N=64:
// element(k4, ntile, lane, e) at idx = k4*256 + ntile*64 + lane*2 + e
// holds W[k4*4 + 2*(lane>>4) + e][ntile*16 + (lane&15)]  (0 if col >= 50).
// Reader: one coalesced b64 per fragment, branch-free.
// ---------------------------------------------------------------------------
__global__ void SOT_wswz_kernel(const float* __restrict__ W,
                                float* __restrict__ Wswz, int K) {
  int idx = blockIdx.x * blockDim.x + threadIdx.x;
  if (idx >= K * FPAD) return;
  int e     = idx & 1;
  int lane  = (idx >> 1) & 31;
  int ntile = (idx >> 6) & 3;
  int k4    = idx >> 8;
  int kh = lane >> 4, pp = lane & 15;
  int n  = ntile * 16 + pp;
  int kk = k4 * 4 + 2 * kh + e;
  Wswz[idx] = (n < FEAT) ? W[(size_t)kk * FEAT + n] : 0.f;
}

// ---------------------------------------------------------------------------
// Zero-pad ir [B,36,64] -> irp [B,48,64] so Cxy B-fragments load branch-free.
// ---------------------------------------------------------------------------
__global__ void SOT_irpad_kernel(const float* __restrict__ ir,
                                 float* __restrict__ irp) {
  int idx = blockIdx.x * blockDim.x + threadIdx.x;  // BATCH*RPAD*FPAD
  if (idx >= BATCH * RPAD * FPAD) return;
  int k = idx & (FPAD - 1);
  int j = (idx >> 6) % RPAD;
  int b = idx / (RPAD * FPAD);
  irp[idx] = (j < RREG) ? ir[((size_t)b * RREG + j) * FPAD + k] : 0.f;
}

// ---------------------------------------------------------------------------
// out[r, j] = j<FEAT ? relu(sum_k X[r,k]*W[k,j] + bias[j]) : 0   (ld = FPAD)
// 128 threads = 4 waves; wave = N-tile; 16 M-rows per block.
// A staged in LDS (stride 68: conflict-free b64 reads, b128-aligned stores);
// B from pre-swizzled Wswz (coalesced b64); 2 accumulators break WMMA chain.
// ---------------------------------------------------------------------------
#define AS_STRIDE 68
__global__ __launch_bounds__(128) void SOT_proj_kernel(
    const float* __restrict__ X, const v2f* __restrict__ Wswz,
    const float* __restrict__ bias, float* __restrict__ out, int K) {
  __shared__ float As[16 * AS_STRIDE];
  const int tid  = threadIdx.x;
  const int lane = tid & 31;
  const int wave = tid >> 5;       // N tile 0..3
  const int kh   = lane >> 4;      // K pair select
  const int p    = lane & 15;      // M (A) / N (B, C)
  const int m0   = blockIdx.x * 16;
  const int n    = wave * 16 + p;

  v8f acc0 = V8F_ZERO, acc1 = V8F_ZERO;
  for (int k0 = 0; k0 < K; k0 += 64) {
    for (int idx = tid; idx < 16 * 16; idx += 128) {   // 256 float4 fills
      int r = idx >> 4, c4 = (idx & 15) << 2;
      float4 v = *(const float4*)(X + (size_t)(m0 + r) * K + k0 + c4);
      *(float4*)(&As[r * AS_STRIDE + c4]) = v;
    }
    if (k0 + 64 < K)
      __builtin_prefetch(X + (size_t)(m0 + (tid >> 3)) * K + k0 + 64 + ((tid & 7) << 3), 0, 3);
    __syncthreads();
    const v2f* Wk = Wswz + ((size_t)(k0 >> 2) * 4 + wave) * 32 + lane;
    for (int kk = 0; kk < 64; kk += 8) {
      v2f a0 = *(const v2f*)(&As[p * AS_STRIDE + kk + 2 * kh]);
      v2f b0 = Wk[(kk >> 2) * 128];
      acc0 = wmma4(a0, b0, acc0);
      v2f a1 = *(const v2f*)(&As[p * AS_STRIDE + kk + 4 + 2 * kh]);
      v2f b1 = Wk[((kk >> 2) + 1) * 128];
      acc1 = wmma4(a1, b1, acc1);
    }
    __syncthreads();
  }
  v8f acc = acc0 + acc1;
  float bj = (n < FEAT) ? bias[n] : 0.0f;
  for (int v = 0; v < 8; ++v) {
    int row = m0 + v + 8 * kh;
    float val = (n < FEAT) ? fmaxf(acc[v] + bj, 0.0f) : 0.0f;
    out[(size_t)row * FPAD + n] = val;
  }
}

// ---------------------------------------------------------------------------
__global__ void SOT_sqnorm_kernel(const float* __restrict__ Xp,
                                  float* __restrict__ nrm, int nrows) {
  int r = blockIdx.x * blockDim.x + threadIdx.x;
  if (r >= nrows) return;
  const float* p = Xp + (size_t)r * FPAD;
  float s = 0.f;
  for (int j = 0; j < FEAT; ++j) { float v = p[j]; s += v * v; }
  nrm[r] = s;
}

// ---------------------------------------------------------------------------
__global__ __launch_bounds__(256) void SOT_nwords_kernel(
    const int* __restrict__ mask, float* __restrict__ nw) {
  __shared__ float red[256];
  int b = blockIdx.x;
  float v = (float)mask[(size_t)b * LTOK + threadIdx.x];
  float s = block_reduce_sum_256(v, red);
  if (threadIdx.x == 0) nw[b] = s;
}

// ---------------------------------------------------------------------------
// Small MLP head -> mix_pred[b,2]. One block per batch row.
// ---------------------------------------------------------------------------
__global__ __launch_bounds__(256) void SOT_mlp_kernel(
    const float* __restrict__ txt_global, const float* __restrict__ img_global,
    const float* __restrict__ social,
    const float* __restrict__ W_stat, const float* __restrict__ b_stat,
    const float* __restrict__ W_gt, const float* __restrict__ b_gt,
    const float* __restrict__ W_gi, const float* __restrict__ b_gi,
    const float* __restrict__ W_m1, const float* __restrict__ b_m1,
    const float* __restrict__ W_m2, const float* __restrict__ b_m2,
    float* __restrict__ mix_out) {
  __shared__ float soc[100];
  __shared__ float tgig[200];
  __shared__ float h[100];
  int b = blockIdx.x, tid = threadIdx.x;
  if (tid < 100) {
    float s = b_stat[tid];
    for (int k = 0; k < DSOC; ++k) s += social[b * DSOC + k] * W_stat[k * 100 + tid];
    soc[tid] = fmaxf(s, 0.f);
  }
  __syncthreads();
  if (tid < 200) {
    float s = b_gt[tid];
    for (int k = 0; k < DTXT; ++k) s += txt_global[(size_t)b * DTXT + k] * W_gt[(size_t)k * 200 + tid];
    for (int k = 0; k < 100; ++k) s += soc[k] * W_gt[(size_t)(DTXT + k) * 200 + tid];
    float tg = fmaxf(s, 0.f);
    float si = b_gi[tid];
    for (int k = 0; k < DIMG; ++k) si += img_global[(size_t)b * DIMG + k] * W_gi[(size_t)k * 200 + tid];
    tgig[tid] = tg + fmaxf(si, 0.f);
  }
  __syncthreads();
  if (tid < 100) {
    float s = b_m1[tid];
    for (int k = 0; k < 200; ++k) s += tgig[k] * W_m1[k * 100 + tid];
    h[tid] = fmaxf(s, 0.f);
  }
  __syncthreads();
  if (tid < 2) {
    float s = b_m2[tid];
    for (int k = 0; k < 100; ++k) s += h[k] * W_m2[k * 2 + tid];
    mix_out[b * 2 + tid] = s;
  }
}

// ---------------------------------------------------------------------------
// Sxy: C/eps (256x36) built in LDS with WMMA (B from padded irp), then 20
// Sinkhorn iterations with uniform loop bound nvalid (mask is a prefix) and
// two-pass logsumexp (1 exp / element). Stride 37 -> conflict-free columns.
// ---------------------------------------------------------------------------
#define XY_STRIDE 37
__global__ __launch_bounds__(256) void SOT_sinkhorn_xy_kernel(
    const float* __restrict__ tr, const float* __restrict__ irp,
    const float* __restrict__ xx, const float* __restrict__ yy,
    const float* __restrict__ nw, float* __restrict__ Sxy) {
  __shared__ float C[LTOK * XY_STRIDE];
  __shared__ float fs[LTOK];
  __shared__ float gs[RREG];
  __shared__ float red[256];
  const int b = blockIdx.x, tid = threadIdx.x;
  const int lane = tid & 31, wave = tid >> 5;
  const int kh = lane >> 4, p = lane & 15;
  const float* trb = tr + (size_t)b * LTOK * FPAD;
  const float* irb = irp + (size_t)b * RPAD * FPAD;
  const float* xxb = xx + (size_t)b * LTOK;
  const float* yyb = yy + (size_t)b * RREG;

  for (int t = wave; t < 16 * 3; t += 8) {   // 16 M-tiles x 3 N-tiles
    int tm = t / 3, tn = t % 3;
    int n = tn * 16 + p;
    v8f acc0 = V8F_ZERO, acc1 = V8F_ZERO;
    const float* ar = trb + (tm * 16 + p) * FPAD;
    const float* br = irb + n * FPAD;
    for (int kk = 0; kk < FPAD; kk += 8) {
      int ka = kk + 2 * kh;
      acc0 = wmma4(*(const v2f*)(ar + ka), *(const v2f*)(br + ka), acc0);
      acc1 = wmma4(*(const v2f*)(ar + ka + 4), *(const v2f*)(br + ka + 4), acc1);
    }
    v8f acc = acc0 + acc1;
    if (n < RREG) {
      float yj = yyb[n];
      for (int v = 0; v < 8; ++v) {
        int i = tm * 16 + v + 8 * kh;
        C[i * XY_STRIDE + n] = (0.5f * xxb[i] + 0.5f * yj - acc[v]) * INV_EPS;
      }
    }
  }
  const int nvalid = (int)(nw[b] + 0.5f);
  const float la = -__logf((float)nvalid);
  const float lb = -__logf((float)RREG);
  fs[tid] = 0.f;
  if (tid < RREG) gs[tid] = 0.f;
  __syncthreads();

  for (int it = 0; it < NITER; ++it) {
    if (tid < RREG) {                 // g_j = -(la + lse_{i<nv}(fs_i - Cs_ij))
      const int j = tid;
      float m = NEG_INF;
      for (int i = 0; i < nvalid; ++i) m = fmaxf(m, fs[i] - C[i * XY_STRIDE + j]);
      float s = 0.f;
      for (int i = 0; i < nvalid; ++i) s += __expf(fs[i] - C[i * XY_STRIDE + j] - m);
      gs[j] = -(la + m + __logf(s));
    }
    __syncthreads();
    {                                  // f_i = -(lb + lse_j(gs_j - Cs_ij))
      const int i = tid;
      float m = NEG_INF;
      for (int j = 0; j < RREG; ++j) m = fmaxf(m, gs[j] - C[i * XY_STRIDE + j]);
      float s = 0.f;
      for (int j = 0; j < RREG; ++j) s += __expf(gs[j] - C[i * XY_STRIDE + j] - m);
      fs[i] = -(lb + m + __logf(s));
    }
    __syncthreads();
  }
  float contrib = 0.f;
  if (tid < nvalid) contrib += (1.0f / (float)nvalid) * (EPSV * fs[tid]);
  if (tid < RREG)   contrib += (1.0f / (float)RREG) * (EPSV * gs[tid]);
  float tot = block_reduce_sum_256(contrib, red);
  if (tid == 0) Sxy[b] = tot;
}

// ---------------------------------------------------------------------------
// Sxx: 256x256 cost matrix resident in dynamic LDS (267 KB; CDNA5 has 320 KB).
// ---------------------------------------------------------------------------
#define XX_STRIDE 257
__global__ __launch_bounds__(256) void SOT_sinkhorn_xx_kernel(
    const float* __restrict__ tr, const float* __restrict__ xx,
    const float* __restrict__ nw, float* __restrict__ Sxx) {
  extern __shared__ float sm[];
  float* C   = sm;                        // 256*257
  float* fs  = C + LTOK * XX_STRIDE;      // 256
  float* gs  = fs + LTOK;                 // 256
  float* red = gs + LTOK;                 // 256
  const int b = blockIdx.x, tid = threadIdx.x;
  const int lane = tid & 31, wave = tid >> 5;
  const int kh = lane >> 4, p = lane & 15;
  const float* trb = tr + (size_t)b * LTOK * FPAD;
  const float* xxb = xx + (size_t)b * LTOK;

  for (int t = wave; t < 256; t += 8) {     // 16x16 tile grid
    int tm = t >> 4, tn = t & 15;
    int n = tn * 16 + p;
    v8f acc0 = V8F_ZERO, acc1 = V8F_ZERO;
    const float* ar = trb + (tm * 16 + p) * FPAD;
    const float* br = trb + n * FPAD;
    for (int kk = 0; kk < FPAD; kk += 8) {
      int ka = kk + 2 * kh;
      acc0 = wmma4(*(const v2f*)(ar + ka), *(const v2f*)(br + ka), acc0);
      acc1 = wmma4(*(const v2f*)(ar + ka + 4), *(const v2f*)(br + ka + 4), acc1);
    }
    v8f acc = acc0 + acc1;
    float xn = xxb[n];
    for (int v = 0; v < 8; ++v) {
      int i = tm * 16 + v + 8 * kh;
      C[i * XX_STRIDE + n] = (0.5f * xxb[i] + 0.5f * xn - acc[v]) * INV_EPS;
    }
  }
  const int nvalid = (int)(nw[b] + 0.5f);
  const float la = -__logf((float)nvalid);
  fs[tid] = 0.f;
  gs[tid] = 0.f;
  __syncthreads();

  for (int it = 0; it < NITER; ++it) {
    if (tid < nvalid) {                // g_j over valid i
      const int j = tid;
      float m = NEG_INF;
      for (int i = 0; i < nvalid; ++i) m = fmaxf(m, fs[i] - C[i * XX_STRIDE + j]);
      float s = 0.f;
      for (int i = 0; i < nvalid; ++i) s += __expf(fs[i] - C[i * XX_STRIDE + j] - m);
      gs[j] = -(la + m + __logf(s));
    }
    __syncthreads();
    if (tid < nvalid) {                // f_i over valid j
      const int i = tid;
      float m = NEG_INF;
      for (int j = 0; j < nvalid; ++j) m = fmaxf(m, gs[j] - C[i * XX_STRIDE + j]);
      float s = 0.f;
      for (int j = 0; j < nvalid; ++j) s += __expf(gs[j] - C[i * XX_STRIDE + j] - m);
      fs[i] = -(la + m + __logf(s));
    }
    __syncthreads();
  }
  float contrib = 0.f;
  if (tid < nvalid)
    contrib = (1.0f / (float)nvalid) * EPSV * (fs[tid] + gs[tid]);
  float tot = block_reduce_sum_256(contrib, red);
  if (tid == 0) Sxx[b] = tot;
}

// ---------------------------------------------------------------------------
// Syy: tiny 36x36; scalar cost build + 20 iterations (two-pass LSE).
// ---------------------------------------------------------------------------
__global__ __launch_bounds__(256) void SOT_sinkhorn_yy_kernel(
    const float* __restrict__ ir, const float* __restrict__ yy,
    float* __restrict__ Syy) {
  __shared__ float C[RREG * RREG];
  __shared__ float fs[RREG];
  __shared__ float gs[RREG];
  __shared__ float red[256];
  const int b = blockIdx.x, tid = threadIdx.x;
  const float* irb = ir + (size_t)b * RREG * FPAD;
  const float* yyb = yy + (size_t)b * RREG;
  for (int e = tid; e < RREG * RREG; e += 256) {
    int i = e / RREG, j = e % RREG;
    float dot = 0.f;
    for (int k = 0; k < FEAT; ++k) dot += irb[i * FPAD + k] * irb[j * FPAD + k];
    C[e] = (0.5f * yyb[i] + 0.5f * yyb[j] - dot) * INV_EPS;
  }
  if (tid < RREG) { fs[tid] = 0.f; gs[tid] = 0.f; }
  __syncthreads();
  const float lb = -__logf((float)RREG);
  for (int it = 0; it < NITER; ++it) {
    if (tid < RREG) {
      const int j = tid;
      float m = NEG_INF;
      for (int i = 0; i < RREG; ++i) m = fmaxf(m, fs[i] - C[i * RREG + j]);
      float s = 0.f;
      for (int i = 0; i < RREG; ++i) s += __expf(fs[i] - C[i * RREG + j] - m);
      gs[j] = -(lb + m + __logf(s));
    }
    __syncthreads();
    if (tid < RREG) {
      const int i = tid;
      float m = NEG_INF;
      for (int j = 0; j < RREG; ++j) m = fmaxf(m, gs[j] - C[i * RREG + j]);
      float s = 0.f;
      for (int j = 0; j < RREG; ++j) s += __expf(gs[j] - C[i * RREG + j] - m);
      fs[i] = -(lb + m + __logf(s));
    }
    __syncthreads();
  }
  float contrib = (tid < RREG) ? (1.0f / (float)RREG) * EPSV * (fs[tid] + gs[tid]) : 0.f;
  float tot = block_reduce_sum_256(contrib, red);
  if (tid == 0) Syy[b] = tot;
}

// ---------------------------------------------------------------------------
__global__ void SOT_finalize_kernel(const float* __restrict__ mix,
                                    const float* __restrict__ Sxy,
                                    const float* __restrict__ Sxx,
                                    const float* __restrict__ Syy,
                                    float* __restrict__ out) {
  int b = blockIdx.x * blockDim.x + threadIdx.x;
  if (b >= BATCH) return;
  float wdis = Sxy[b] - 0.5f * (Sxx[b] + Syy[b]);
  float wp0 = 1.0f - GAMMA_ * wdis, wp1 = GAMMA_ * wdis;
  float z0 = fmaxf(mix[b * 2 + 0], wp0);
  float z1 = fmaxf(mix[b * 2 + 1], wp1);
  float m = fmaxf(z0, z1);
  float e0 = __expf(z0 - m), e1 = __expf(z1 - m);
  float inv = 1.0f / (e0 + e1);
  out[b * 2 + 0] = e0 * inv;
  out[b * 2 + 1] = e1 * inv;
}

// ---------------------------------------------------------------------------
extern "C" void kernel_launch(void* const* d_in, const int* in_sizes, int n_in,
                              void* d_out, int out_size, void* d_ws, size_t ws_size,
                              hipStream_t stream) {
  (void)in_sizes; (void)n_in; (void)out_size; (void)ws_size;
  const float* txt_global = (const float*)d_in[0];
  const float* txt_region = (const float*)d_in[1];
  const float* img_global = (const float*)d_in[2];
  const float* img_region = (const float*)d_in[3];
  const float* social     = (const float*)d_in[4];
  const int*   attn_mask  = (const int*)d_in[5];
  const float* W_stat = (const float*)d_in[6];
  const float* b_stat = (const float*)d_in[7];
  const float* W_gt   = (const float*)d_in[8];
  const float* b_gt   = (const float*)d_in[9];
  const float* W_gi   = (const float*)d_in[10];
  const float* b_gi   = (const float*)d_in[11];
  const float* W_rt   = (const float*)d_in[12];
  const float* b_rt   = (const float*)d_in[13];
  const float* W_ri   = (const float*)d_in[14];
  const float* b_ri   = (const float*)d_in[15];
  const float* W_m1   = (const float*)d_in[16];
  const float* b_m1   = (const float*)d_in[17];
  const float* W_m2   = (const float*)d_in[18];
  const float* b_m2   = (const float*)d_in[19];
  float* out = (float*)d_out;

  const int TR_ROWS = BATCH * LTOK;   // 32768
  const int IR_ROWS = BATCH * RREG;   // 4608

  float* ws    = (float*)d_ws;
  float* tr    = ws;
  float* ir    = tr + (size_t)TR_ROWS * FPAD;
  float* irp   = ir + (size_t)IR_ROWS * FPAD;
  float* wrt_s = irp + (size_t)BATCH * RPAD * FPAD;
  float* wri_s = wrt_s + (size_t)DTXT * FPAD;
  float* xx    = wri_s + (size_t)DIMG * FPAD;
  float* yy    = xx + TR_ROWS;
  float* nw    = yy + IR_ROWS;
  float* mix   = nw + BATCH;
  float* Sxy   = mix + BATCH * 2;
  float* Sxx   = Sxy + BATCH;
  float* Syy   = Sxx + BATCH;

  SOT_wswz_kernel<<<(DTXT * FPAD + 255) / 256, 256, 0, stream>>>(W_rt, wrt_s, DTXT);
  SOT_wswz_kernel<<<(DIMG * FPAD + 255) / 256, 256, 0, stream>>>(W_ri, wri_s, DIMG);

  SOT_mlp_kernel<<<BATCH, 256, 0, stream>>>(
      txt_global, img_global, social, W_stat, b_stat, W_gt, b_gt,
      W_gi, b_gi, W_m1, b_m1, W_m2, b_m2, mix);
  SOT_nwords_kernel<<<BATCH, 256, 0, stream>>>(attn_mask, nw);

  SOT_proj_kernel<<<TR_ROWS / 16, 128, 0, stream>>>(txt_region, (const v2f*)wrt_s, b_rt, tr, DTXT);
  SOT_proj_kernel<<<IR_ROWS / 16, 128, 0, stream>>>(img_region, (const v2f*)wri_s, b_ri, ir, DIMG);

  SOT_irpad_kernel<<<(BATCH * RPAD * FPAD + 255) / 256, 256, 0, stream>>>(ir, irp);
  SOT_sqnorm_kernel<<<(TR_ROWS + 255) / 256, 256, 0, stream>>>(tr, xx, TR_ROWS);
  SOT_sqnorm_kernel<<<(IR_ROWS + 255) / 256, 256, 0, stream>>>(ir, yy, IR_ROWS);

  SOT_sinkhorn_xy_kernel<<<BATCH, 256, 0, stream>>>(tr, irp, xx, yy, nw, Sxy);
  size_t xx_lds = (size_t)(LTOK * XX_STRIDE + 3 * LTOK) * sizeof(float); // ~267 KB
  SOT_sinkhorn_xx_kernel<<<BATCH, 256, xx_lds, stream>>>(tr, xx, nw, Sxx);
  SOT_sinkhorn_yy_kernel<<<BATCH, 256, 0, stream>>>(ir, yy, Syy);

  SOT_finalize_kernel<<<1, 128, 0, stream>>>(mix, Sxy, Sxx, Syy, out);
}